// FrameAlignedPointError_28784870817856
// MI455X (gfx1250) — compile-verified
//
#include <hip/hip_runtime.h>

// ---------------------------------------------------------------------------
// FAPE loss, MI455X (gfx1250, wave32).
// dist[b,f,k] = ||R_f (d_k - d_f)|| = ||d_k - d_f||  (R_f orthonormal)
//             = sqrt(||d_k||^2 + ||d_f||^2 - 2 d_k.d_f + eps)
// Cross term d_k . d_f computed with V_WMMA_F32_16X16X4_F32 tiles.
// ---------------------------------------------------------------------------

typedef __attribute__((ext_vector_type(2))) float v2f;
typedef __attribute__((ext_vector_type(8))) float v8f;

#define B_DIM      64
#define N_DIM      4096
#define NFRAMES    100
#define FPAD       112   // 7 tiles of 16
#define FTILES     7
#define CLAMP_D    10.0f
#define INV_UNIT   0.1f
#define EPS_DIST   1e-8f

__global__ void fape_zero(double* acc) {
    acc[0] = 0.0;   // error sum
    acc[1] = 0.0;   // mask sum
}

// d4[i] = pred[i]-true[i] padded to float4 (w=0); nk[i] = ||d||^2; mask sum -> acc[1]
__global__ __launch_bounds__(256)
void fape_prep(const float* __restrict__ pred, const float* __restrict__ tru,
               const float* __restrict__ mask, float* __restrict__ d4,
               float* __restrict__ nk, double* __restrict__ acc, int total) {
    int i = blockIdx.x * blockDim.x + threadIdx.x;
    float m = 0.0f;
    if (i < total) {
        float dx = pred[3 * i + 0] - tru[3 * i + 0];
        float dy = pred[3 * i + 1] - tru[3 * i + 1];
        float dz = pred[3 * i + 2] - tru[3 * i + 2];
        d4[4 * i + 0] = dx;
        d4[4 * i + 1] = dy;
        d4[4 * i + 2] = dz;
        d4[4 * i + 3] = 0.0f;
        nk[i] = dx * dx + dy * dy + dz * dz;
        m = mask[i];
    }
    __shared__ float red[256];
    red[threadIdx.x] = m;
    __syncthreads();
    for (int s = 128; s > 0; s >>= 1) {
        if (threadIdx.x < (unsigned)s) red[threadIdx.x] += red[threadIdx.x + s];
        __syncthreads();
    }
    if (threadIdx.x == 0) atomicAdd(&acc[1], (double)red[0]);
}

// grid = (N/128, B); block = 256 (8 waves). Each wave owns one 16-point tile
// and sweeps 7 frame tiles with wmma_f32_16x16x4_f32.
__global__ __launch_bounds__(256)
void fape_main(const float* __restrict__ d4, const float* __restrict__ nk,
               const float* __restrict__ mask, double* __restrict__ acc) {
    const int b = blockIdx.y;

    __shared__ float s_of[FPAD * 4];  // frame deltas (x,y,z,0)
    __shared__ float s_on[FPAD];      // frame ||d_f||^2
    for (int f = threadIdx.x; f < FPAD; f += blockDim.x) {
        if (f < NFRAMES) {
            const float* p = d4 + (size_t)(b * N_DIM + f) * 4;
            s_of[4 * f + 0] = p[0];
            s_of[4 * f + 1] = p[1];
            s_of[4 * f + 2] = p[2];
            s_of[4 * f + 3] = 0.0f;
            s_on[f] = nk[b * N_DIM + f];
        } else {
            s_of[4 * f + 0] = 0.0f;
            s_of[4 * f + 1] = 0.0f;
            s_of[4 * f + 2] = 0.0f;
            s_of[4 * f + 3] = 0.0f;
            s_on[f] = 0.0f;
        }
    }
    __syncthreads();

    const int wave = threadIdx.x >> 5;
    const int lane = threadIdx.x & 31;
    const int half = lane >> 4;    // 0: lanes 0-15, 1: lanes 16-31
    const int col  = lane & 15;
    const int k0   = (blockIdx.x * 8 + wave) * 16;

    // A operand (16x4 f32): lane holds A[col][2*half + v], v = 0,1
    const size_t abase = ((size_t)(b * N_DIM + k0 + col)) * 4 + 2 * half;
    v2f A;
    A.x = d4[abase + 0];
    A.y = d4[abase + 1];   // component 3 is the stored zero pad

    // Row-side norms and mask for this lane's 8 C-rows (M = r + 8*half)
    float rn[8], rm[8];
#pragma unroll
    for (int r = 0; r < 8; ++r) {
        int k = k0 + r + 8 * half;
        rn[r] = nk[b * N_DIM + k];
        rm[r] = mask[b * N_DIM + k];
    }

    float lsum = 0.0f;
#pragma unroll
    for (int ft = 0; ft < FTILES; ++ft) {
        const int f0 = ft * 16;
        // B operand (4x16 f32): lane holds B[2*half + v][col] = d_f[f0+col][comp]
        const int fb = (f0 + col) * 4 + 2 * half;
        v2f Bv;
        Bv.x = s_of[fb + 0];
        Bv.y = s_of[fb + 1];

        v8f C = {};
        C = __builtin_amdgcn_wmma_f32_16x16x4_f32(
                /*neg_a=*/false, A, /*neg_b=*/false, Bv,
                /*c_mod=*/(short)0, C, /*reuse_a=*/false, /*reuse_b=*/false);

        const float on   = s_on[f0 + col];
        const float fval = (f0 + col < NFRAMES) ? 1.0f : 0.0f;
#pragma unroll
        for (int r = 0; r < 8; ++r) {
            float d2 = rn[r] + on - 2.0f * C[r];
            d2 = fmaxf(d2, 0.0f) + EPS_DIST;
            float dist = sqrtf(d2);
            lsum += fminf(dist, CLAMP_D) * rm[r] * fval;
        }
    }
    lsum *= INV_UNIT;

    __shared__ float red[256];
    red[threadIdx.x] = lsum;
    __syncthreads();
    for (int s = 128; s > 0; s >>= 1) {
        if (threadIdx.x < (unsigned)s) red[threadIdx.x] += red[threadIdx.x + s];
        __syncthreads();
    }
    if (threadIdx.x == 0) atomicAdd(&acc[0], (double)red[0]);
}

__global__ void fape_final(const double* __restrict__ acc, float* __restrict__ out) {
    out[0] = (float)(acc[0] / (acc[1] + 1e-8));
}

extern "C" void kernel_launch(void* const* d_in, const int* in_sizes, int n_in,
                              void* d_out, int out_size, void* d_ws, size_t ws_size,
                              hipStream_t stream) {
    const float* pred = (const float*)d_in[0];
    const float* tru  = (const float*)d_in[1];
    const float* mask = (const float*)d_in[2];
    float* out = (float*)d_out;

    const int total = B_DIM * N_DIM;  // 262144

    char* ws = (char*)d_ws;
    float*  d4  = (float*)ws;                                          // total*4 floats = 4 MB
    float*  nkv = (float*)(ws + (size_t)total * 4 * sizeof(float));    // total floats   = 1 MB
    double* acc = (double*)(ws + (size_t)total * 5 * sizeof(float));   // 2 doubles (8B aligned)

    fape_zero<<<1, 1, 0, stream>>>(acc);
    fape_prep<<<(total + 255) / 256, 256, 0, stream>>>(pred, tru, mask, d4, nkv, acc, total);

    dim3 grid(N_DIM / 128, B_DIM);  // 32 x 64 blocks, 8 waves each
    fape_main<<<grid, 256, 0, stream>>>(d4, nkv, mask, acc);

    fape_final<<<1, 1, 0, stream>>>(acc, out);
}